// HouseholdAssignmentGNN_8770323218930
// MI455X (gfx1250) — compile-verified
//
#include <hip/hip_runtime.h>

typedef __attribute__((ext_vector_type(2))) float v2f;
typedef __attribute__((ext_vector_type(8))) float v8f;

// ---------------------------------------------------------------------------
// Fused dual GEMM (one pass over the activations):
//   y = in @ Wl              (columns handled by waves 0,1)
//   z = in @ Wr + bias       (columns handled by waves 2,3)
// in: [N,K] row-major, Wl/Wr: [K,32] row-major, y/z: [N,32].
// One wave computes one 16x16 output tile via V_WMMA_F32_16X16X4_F32.
// N assumed multiple of 16 (100000 = 6250*16).
// ---------------------------------------------------------------------------
__global__ __launch_bounds__(128)
void gemm_dual_wmma(const float* __restrict__ in, int K,
                    const float* __restrict__ Wl, const float* __restrict__ Wr,
                    const float* __restrict__ bias,
                    float* __restrict__ y, float* __restrict__ z) {
  const int lane = threadIdx.x & 31;
  const int wave = threadIdx.x >> 5;        // 0..3
  const int row0 = blockIdx.x * 16;
  const int m    = lane & 15;               // A row / B col / C col within tile
  const int kh   = lane >> 4;               // selects K pair within 4-wide step
  const int ncol = ((wave & 1) << 4) | m;   // column in the 32-wide output
  const float* W    = (wave < 2) ? Wl : Wr;
  const float* arow = in + (size_t)(row0 + m) * (size_t)K;

  v8f acc = {};
  const int ksteps = K >> 2;
  for (int kk = 0; kk < ksteps; ++kk) {
    const int kb = (kk << 2) + (kh << 1);
    v2f a, b;
    a.x = arow[kb];
    a.y = arow[kb + 1];
    b.x = W[kb * 32 + ncol];
    b.y = W[(kb + 1) * 32 + ncol];
    acc = __builtin_amdgcn_wmma_f32_16x16x4_f32(false, a, false, b,
                                                (short)0, acc, false, false);
  }

  const float bv   = (wave >= 2) ? bias[ncol] : 0.0f;
  float* dst       = (wave < 2) ? y : z;
#pragma unroll
  for (int v = 0; v < 8; ++v) {
    const int r = row0 + (kh << 3) + v;     // C row = v + 8*(lane/16)
    dst[(size_t)r * 32 + ncol] = acc[v] + bv;
  }
}

// ---------------------------------------------------------------------------
// Degree: one thread per edge (graph is identical for both layers).
// ---------------------------------------------------------------------------
__global__ void deg_kernel(const int* __restrict__ dst, float* __restrict__ deg,
                           int E) {
  const int e = blockIdx.x * blockDim.x + threadIdx.x;
  if (e < E) atomicAdd(&deg[dst[e]], 1.0f);
}

// ---------------------------------------------------------------------------
// Edge aggregation: one wave per edge, lane = channel (coalesced 128B rows).
// agg (12.8MB) and feat (12.8MB) both fit in the 192MB L2, so the atomics
// stay on-die.
// ---------------------------------------------------------------------------
__global__ __launch_bounds__(256)
void agg_kernel(const int* __restrict__ src, const int* __restrict__ dst,
                const float* __restrict__ feat, float* __restrict__ agg, int E) {
  const int e = blockIdx.x * 8 + (threadIdx.x >> 5);
  const int c = threadIdx.x & 31;
  if (e < E) {
    const int s = src[e];
    const int d = dst[e];
    atomicAdd(&agg[(size_t)d * 32 + c], feat[(size_t)s * 32 + c]);
  }
}

// ---------------------------------------------------------------------------
// h = relu(agg / max(deg,1) + z), elementwise over N*32.
// ---------------------------------------------------------------------------
__global__ __launch_bounds__(256)
void combine_kernel(const float* __restrict__ agg, const float* __restrict__ deg,
                    const float* __restrict__ z, float* __restrict__ h,
                    int total) {
  const int i = blockIdx.x * blockDim.x + threadIdx.x;
  if (i < total) {
    float dg = deg[i >> 5];
    dg = dg > 1.0f ? dg : 1.0f;
    const float v = agg[i] / dg + z[i];
    h[i] = v > 0.0f ? v : 0.0f;
  }
}

// ---------------------------------------------------------------------------
// Output GEMM: out = h @ Wfc + bfc.  h: [N,32], Wfc: [32,NCOL], NCOL=2000.
// 8 waves/block; each wave preloads its 8 A-fragments (16 regs) once and
// sweeps column tiles ct = wave, wave+8, ... (125 tiles). Wfc (256KB) is
// L2-resident; 800MB output streams at HBM rate.
// ---------------------------------------------------------------------------
__global__ __launch_bounds__(256)
void gemm_out_wmma(const float* __restrict__ h, const float* __restrict__ Wfc,
                   const float* __restrict__ bfc, float* __restrict__ out,
                   int NCOL) {
  const int lane = threadIdx.x & 31;
  const int wave = threadIdx.x >> 5;        // 0..7
  const int row0 = blockIdx.x * 16;
  const int m    = lane & 15;
  const int kh   = lane >> 4;

  const float* arow = h + (size_t)(row0 + m) * 32;
  v2f afrag[8];
#pragma unroll
  for (int kk = 0; kk < 8; ++kk) {
    const int kb = (kk << 2) + (kh << 1);
    afrag[kk].x = arow[kb];
    afrag[kk].y = arow[kb + 1];
  }

  const int ntiles = NCOL >> 4;             // 125
  for (int ct = wave; ct < ntiles; ct += 8) {
    const int col = (ct << 4) + m;
    v8f acc = {};
#pragma unroll
    for (int kk = 0; kk < 8; ++kk) {
      const int kb = (kk << 2) + (kh << 1);
      v2f b;
      b.x = Wfc[kb * NCOL + col];
      b.y = Wfc[(kb + 1) * NCOL + col];
      acc = __builtin_amdgcn_wmma_f32_16x16x4_f32(false, afrag[kk], false, b,
                                                  (short)0, acc, false, false);
    }
    const float bv = bfc[col];
#pragma unroll
    for (int v = 0; v < 8; ++v) {
      const int r = row0 + (kh << 3) + v;
      out[(size_t)r * NCOL + col] = acc[v] + bv;
    }
  }
}

// ---------------------------------------------------------------------------
// Launch: project-then-aggregate (mean aggregation commutes with the linear
// layer), cutting edge traffic 4x vs aggregating 128-ch features.
// ---------------------------------------------------------------------------
extern "C" void kernel_launch(void* const* d_in, const int* in_sizes, int n_in,
                              void* d_out, int out_size, void* d_ws,
                              size_t ws_size, hipStream_t stream) {
  const float* x    = (const float*)d_in[0];
  const int*   ei   = (const int*)d_in[1];
  const float* W1l  = (const float*)d_in[2];
  const float* b1   = (const float*)d_in[3];
  const float* W1r  = (const float*)d_in[4];
  const float* W3l  = (const float*)d_in[5];
  const float* b3   = (const float*)d_in[6];
  const float* W3r  = (const float*)d_in[7];
  const float* Wfc  = (const float*)d_in[8];
  const float* bfc  = (const float*)d_in[9];
  float* out = (float*)d_out;

  const int N    = in_sizes[0] / 128;   // 100000
  const int E    = in_sizes[1] / 2;     // 3200000
  const int NCOL = in_sizes[9];         // 2000

  const int* src = ei;                  // edge_index[0]
  const int* dstv = ei + E;             // edge_index[1]

  float* ws = (float*)d_ws;
  const size_t nf = (size_t)N * 32;
  float* y   = ws;            // projected messages          [N,32]
  float* z   = y   + nf;      // self-term (+bias)           [N,32]
  float* agg = z   + nf;      // neighbor sums               [N,32]
  float* h1  = agg + nf;      // layer-1 activations         [N,32]
  float* h2  = h1  + nf;      // layer-2 activations         [N,32]
  float* deg = h2  + nf;      // in-degrees                  [N]

  const int rowTiles = N / 16;          // 6250 (N divisible by 16)

  // ---- Layer 1 ----
  gemm_dual_wmma<<<rowTiles, 128, 0, stream>>>(x, 128, W1l, W1r, b1, y, z);
  hipMemsetAsync(deg, 0, (size_t)N * sizeof(float), stream);
  hipMemsetAsync(agg, 0, nf * sizeof(float), stream);
  deg_kernel<<<(E + 255) / 256, 256, 0, stream>>>(dstv, deg, E);
  agg_kernel<<<(E + 7) / 8, 256, 0, stream>>>(src, dstv, y, agg, E);
  combine_kernel<<<((int)nf + 255) / 256, 256, 0, stream>>>(agg, deg, z, h1,
                                                            (int)nf);

  // ---- Layer 2 (same graph, so deg is reused) ----
  gemm_dual_wmma<<<rowTiles, 128, 0, stream>>>(h1, 32, W3l, W3r, b3, y, z);
  hipMemsetAsync(agg, 0, nf * sizeof(float), stream);
  agg_kernel<<<(E + 7) / 8, 256, 0, stream>>>(src, dstv, y, agg, E);
  combine_kernel<<<((int)nf + 255) / 256, 256, 0, stream>>>(agg, deg, z, h2,
                                                            (int)nf);

  // ---- FC head ----
  gemm_out_wmma<<<rowTiles, 256, 0, stream>>>(h2, Wfc, bfc, out, NCOL);
}